// RWKV_RNN_68006512165457
// MI455X (gfx1250) — compile-verified
//
#include <hip/hip_runtime.h>
#include <hip/hip_bf16.h>

#define LNUM 24
#define D    1024
#define HD   4096
#define VOC  50277
#define RPB  128   // rows per block (8 waves x 16 rows)

typedef __attribute__((ext_vector_type(2))) float v2f;
typedef __attribute__((ext_vector_type(8))) float v8f;

// epilogue ops (compile-time folded after inlining)
#define OP_NONE    0
#define OP_SQRELU  1
#define OP_ADDBASE 2
#define OP_GATED   3

// ---- CDNA5 async global->LDS copy (ASYNCcnt-tracked) ----
// VDST = 32-bit LDS byte offset (low 32 bits of the generic shared pointer,
// per flat-LDS addressing); VADDR = 64-bit global addr.
__device__ __forceinline__ void async_g2l_b128(void* lds_dst, const void* gsrc) {
  unsigned lds_off = (unsigned)(unsigned long long)(uintptr_t)lds_dst;
  asm volatile("global_load_async_to_lds_b128 %0, %1, off"
               :: "v"(lds_off), "v"(gsrc) : "memory");
}
__device__ __forceinline__ void async_wait0() {
  asm volatile("s_wait_asynccnt 0x0" ::: "memory");
}

__device__ __forceinline__ float block_reduce_sum(float v, float* red) {
  const int t = threadIdx.x;
  red[t] = v;
  __syncthreads();
  for (int off = 128; off > 0; off >>= 1) {
    if (t < off) red[t] += red[t + off];
    __syncthreads();
  }
  float r = red[0];
  __syncthreads();
  return r;
}

// layernorm of global x[D] into LDS xs[D]; 256 threads, each owns 4 channels
__device__ __forceinline__ void block_ln(const float* __restrict__ xg,
                                         const float* __restrict__ w,
                                         const float* __restrict__ b,
                                         float* xs, float* red) {
  const int t = threadIdx.x;
  float v[4];
  float s = 0.f;
#pragma unroll
  for (int j = 0; j < 4; ++j) { v[j] = xg[t + 256 * j]; s += v[j]; }
  const float mean = block_reduce_sum(s, red) * (1.f / (float)D);
  float s2 = 0.f;
#pragma unroll
  for (int j = 0; j < 4; ++j) { float d = v[j] - mean; s2 += d * d; }
  const float var = block_reduce_sum(s2, red) * (1.f / (float)D);
  const float rstd = rsqrtf(var + 1e-5f);
#pragma unroll
  for (int j = 0; j < 4; ++j) {
    const int i = t + 256 * j;
    xs[i] = (v[j] - mean) * rstd * w[i] + b[i];
  }
  __syncthreads();
}

// One wave computes 16 rows of out = A(rows x K, row-major) @ xvec via
// V_WMMA_F32_16X16X4_F32.  A slot: K = vgpr + 2*(lane>=16), M = lane&15.
// B slot: K = vgpr + 2*(lane>=16); replicated across N so all lanes valid.
__device__ __forceinline__ v8f wave_gemv(const float* __restrict__ A, int K,
                                         const float* __restrict__ xs, int row) {
  const int lane = threadIdx.x & 31;
  const int koff = (lane >> 4) << 1;  // 0 or 2
  const float* __restrict__ ap = A + (size_t)row * (size_t)K + koff;
  const float* __restrict__ bp = xs + koff;
  __builtin_prefetch(ap, 0, 0);  // global_prefetch_b8: warm first lines
  v8f acc = {0.f, 0.f, 0.f, 0.f, 0.f, 0.f, 0.f, 0.f};
#pragma unroll 16
  for (int k = 0; k < K; k += 4) {
    v2f a, bv;
    a.x = ap[k];
    a.y = ap[k + 1];
    bv.x = bp[k];
    bv.y = bp[k + 1];
    acc = __builtin_amdgcn_wmma_f32_16x16x4_f32(false, a, false, bv,
                                                (short)0, acc, false, false);
  }
  return acc;
}

// lane 0 holds rows rbase+0..7 in acc[0..7]; lane 16 holds rows rbase+8..15
__device__ __forceinline__ void wave_store(v8f acc, float* out, int rbase, int op,
                                           const float* base, const float* gate,
                                           int maxrow) {
  const int lane = threadIdx.x & 31;
  if ((lane & 15) == 0) {
    const int r0 = rbase + ((lane >> 4) << 3);
    float res[8];
#pragma unroll
    for (int j = 0; j < 8; ++j) {
      const int r = r0 + j;
      float v = acc[j];
      if (op == OP_SQRELU) { v = fmaxf(v, 0.f); v *= v; }
      else if (op == OP_ADDBASE) { v += base[r]; }
      else if (op == OP_GATED) { v = base[r] + v / (1.f + __expf(-gate[r])); }
      res[j] = v;
    }
    if (r0 + 8 <= maxrow) {  // common path: two b128 stores (32B-aligned dsts)
      *(float4*)(out + r0)     = make_float4(res[0], res[1], res[2], res[3]);
      *(float4*)(out + r0 + 4) = make_float4(res[4], res[5], res[6], res[7]);
    } else {                 // head-kernel tail only
#pragma unroll
      for (int j = 0; j < 8; ++j)
        if (r0 + j < maxrow) out[r0 + j] = res[j];
    }
  }
}

// ---------------- kernels ----------------

__global__ __launch_bounds__(256) void k_embed(const int* __restrict__ token,
                                               const float* __restrict__ emb,
                                               const float* __restrict__ w,
                                               const float* __restrict__ b,
                                               float* __restrict__ xout) {
  __shared__ float red[256];
  const int t = threadIdx.x;
  const float* row = emb + (size_t)token[0] * D;
  float v[4];
  float s = 0.f;
#pragma unroll
  for (int j = 0; j < 4; ++j) { v[j] = row[t + 256 * j]; s += v[j]; }
  const float mean = block_reduce_sum(s, red) * (1.f / (float)D);
  float s2 = 0.f;
#pragma unroll
  for (int j = 0; j < 4; ++j) { float d = v[j] - mean; s2 += d * d; }
  const float rstd = rsqrtf(block_reduce_sum(s2, red) * (1.f / (float)D) + 1e-5f);
#pragma unroll
  for (int j = 0; j < 4; ++j) {
    const int i = t + 256 * j;
    xout[i] = (v[j] - mean) * rstd * w[i] + b[i];
  }
}

// grid (D/RPB, 3): y=0 -> k=kw@xk, y=1 -> v=vw@xv, y=2 -> rpre=rw@xr
__global__ __launch_bounds__(256) void k_att_qkv(
    const float* __restrict__ x, const float* __restrict__ l1w,
    const float* __restrict__ l1b, const float* __restrict__ tmk,
    const float* __restrict__ tmv, const float* __restrict__ tmr,
    const float* __restrict__ s_att, const float* __restrict__ kw,
    const float* __restrict__ vw, const float* __restrict__ rw,
    float* __restrict__ kout, float* __restrict__ vout,
    float* __restrict__ rout, float* __restrict__ st_att) {
  __shared__ float xs[D];
  __shared__ float red[256];
  const int t = threadIdx.x;
  block_ln(x, l1w, l1b, xs, red);
  const int sel = blockIdx.y;
  if (sel == 0 && blockIdx.x == 0) {  // new state row 1 = xn
#pragma unroll
    for (int j = 0; j < 4; ++j) { const int i = t + 256 * j; st_att[i] = xs[i]; }
  }
  const float* tm = (sel == 0) ? tmk : (sel == 1) ? tmv : tmr;
#pragma unroll
  for (int j = 0; j < 4; ++j) {  // each thread rewrites only its own channels
    const int i = t + 256 * j;
    const float m = tm[i];
    xs[i] = xs[i] * m + s_att[i] * (1.f - m);
  }
  __syncthreads();
  const float* A = (sel == 0) ? kw : (sel == 1) ? vw : rw;
  float* out = (sel == 0) ? kout : (sel == 1) ? vout : rout;
  const int rbase = blockIdx.x * RPB + (t >> 5) * 16;
  v8f acc = wave_gemv(A, D, xs, rbase + (t & 15));
  wave_store(acc, out, rbase, OP_NONE, nullptr, nullptr, D);
}

// WKV recurrence (elementwise, recomputed per block) + x += ow @ (r*wkv)
__global__ __launch_bounds__(256) void k_att_out(
    float* x, const float* __restrict__ kv, const float* __restrict__ vv,
    const float* __restrict__ rp, const float* __restrict__ aa,
    const float* __restrict__ bb, const float* __restrict__ pp,
    const float* __restrict__ tf, const float* __restrict__ td,
    const float* __restrict__ ow, float* __restrict__ st_aa,
    float* __restrict__ st_bb, float* __restrict__ st_pp) {
  __shared__ float gs[D];
  const int t = threadIdx.x;
#pragma unroll
  for (int j = 0; j < 4; ++j) {
    const int i = t + 256 * j;
    const float k = kv[i], v = vv[i];
    const float r = 1.f / (1.f + __expf(-rp[i]));
    const float a = aa[i], b = bb[i], p = pp[i];
    const float ww = tf[i] + k;
    const float qq = fmaxf(p, ww);
    const float e1 = __expf(p - qq), e2 = __expf(ww - qq);
    gs[i] = r * ((e1 * a + e2 * v) / (e1 * b + e2));
    const float ww2 = p + td[i];
    const float qq2 = fmaxf(ww2, k);
    const float e1b = __expf(ww2 - qq2), e2b = __expf(k - qq2);
    if (blockIdx.x == 0) {
      st_aa[i] = e1b * a + e2b * v;
      st_bb[i] = e1b * b + e2b;
      st_pp[i] = qq2;
    }
  }
  __syncthreads();
  const int rbase = blockIdx.x * RPB + (t >> 5) * 16;
  v8f acc = wave_gemv(ow, D, gs, rbase + (t & 15));
  wave_store(acc, x, rbase, OP_ADDBASE, x, nullptr, D);
}

// grid 8+32 blocks: first 8 -> rpre2 = frw@fxr (D rows); rest -> kk = sqrelu(fkw@fxk) (H rows)
__global__ __launch_bounds__(256) void k_ffn_rk(
    const float* __restrict__ x, const float* __restrict__ l2w,
    const float* __restrict__ l2b, const float* __restrict__ ftmk,
    const float* __restrict__ ftmr, const float* __restrict__ s_ffn,
    const float* __restrict__ frw, const float* __restrict__ fkw,
    float* __restrict__ rout, float* __restrict__ kkout,
    float* __restrict__ st_ffn) {
  __shared__ float xs[D];
  __shared__ float red[256];
  const int t = threadIdx.x;
  block_ln(x, l2w, l2b, xs, red);
  const bool isR = blockIdx.x < (D / RPB);
  if (blockIdx.x == 0) {  // new state row 0 = xn2
#pragma unroll
    for (int j = 0; j < 4; ++j) { const int i = t + 256 * j; st_ffn[i] = xs[i]; }
  }
  const float* tm = isR ? ftmr : ftmk;
#pragma unroll
  for (int j = 0; j < 4; ++j) {
    const int i = t + 256 * j;
    const float m = tm[i];
    xs[i] = xs[i] * m + s_ffn[i] * (1.f - m);
  }
  __syncthreads();
  const int wave = t >> 5;
  if (isR) {
    const int rbase = blockIdx.x * RPB + wave * 16;
    v8f acc = wave_gemv(frw, D, xs, rbase + (t & 15));
    wave_store(acc, rout, rbase, OP_NONE, nullptr, nullptr, D);
  } else {
    const int rbase = (blockIdx.x - D / RPB) * RPB + wave * 16;
    v8f acc = wave_gemv(fkw, D, xs, rbase + (t & 15));
    wave_store(acc, kkout, rbase, OP_SQRELU, nullptr, nullptr, HD);
  }
}

// x += sigmoid(rpre2) * (fvw @ kk), K = H = 4096 staged in LDS via async copy
__global__ __launch_bounds__(256) void k_ffn_out(float* x,
                                                 const float* __restrict__ kk,
                                                 const float* __restrict__ rp2,
                                                 const float* __restrict__ fvw) {
  __shared__ float ks[HD];
  const int t = threadIdx.x;
#pragma unroll
  for (int j = 0; j < HD / (256 * 4); ++j) {   // 4 x b128 per thread, 16KB total
    const int i = (t + 256 * j) * 4;
    async_g2l_b128(&ks[i], kk + i);
  }
  async_wait0();
  __syncthreads();
  const int rbase = blockIdx.x * RPB + (t >> 5) * 16;
  v8f acc = wave_gemv(fvw, HD, ks, rbase + (t & 15));
  wave_store(acc, x, rbase, OP_GATED, x, rp2, D);
}

// logits = head_w @ ln(x)
__global__ __launch_bounds__(256) void k_head(const float* __restrict__ x,
                                              const float* __restrict__ w,
                                              const float* __restrict__ b,
                                              const float* __restrict__ hw,
                                              float* __restrict__ logits) {
  __shared__ float xs[D];
  __shared__ float red[256];
  const int t = threadIdx.x;
  block_ln(x, w, b, xs, red);
  const int rbase = blockIdx.x * RPB + (t >> 5) * 16;
  int row = rbase + (t & 15);
  if (row > VOC - 1) row = VOC - 1;  // clamp loads, guard stores below
  v8f acc = wave_gemv(hw, D, xs, row);
  wave_store(acc, logits, rbase, OP_NONE, nullptr, nullptr, VOC);
}

extern "C" void kernel_launch(void* const* d_in, const int* in_sizes, int n_in,
                              void* d_out, int out_size, void* d_ws, size_t ws_size,
                              hipStream_t stream) {
  (void)in_sizes; (void)n_in; (void)out_size; (void)ws_size;
  const int*   token      = (const int*)d_in[0];
  const float* state      = (const float*)d_in[1];
  const float* emb        = (const float*)d_in[2];
  const float* ln0_w      = (const float*)d_in[3];
  const float* ln0_b      = (const float*)d_in[4];
  const float* ln1_w      = (const float*)d_in[5];
  const float* ln1_b      = (const float*)d_in[6];
  const float* att_tmk    = (const float*)d_in[7];
  const float* att_tmv    = (const float*)d_in[8];
  const float* att_tmr    = (const float*)d_in[9];
  const float* time_first = (const float*)d_in[10];
  const float* time_decay = (const float*)d_in[11];
  const float* att_kw     = (const float*)d_in[12];
  const float* att_vw     = (const float*)d_in[13];
  const float* att_rw     = (const float*)d_in[14];
  const float* att_ow     = (const float*)d_in[15];
  const float* ln2_w      = (const float*)d_in[16];
  const float* ln2_b      = (const float*)d_in[17];
  const float* ffn_tmk    = (const float*)d_in[18];
  const float* ffn_tmr    = (const float*)d_in[19];
  const float* ffn_kw     = (const float*)d_in[20];
  const float* ffn_vw     = (const float*)d_in[21];
  const float* ffn_rw     = (const float*)d_in[22];
  const float* lnout_w    = (const float*)d_in[23];
  const float* lnout_b    = (const float*)d_in[24];
  const float* head_w     = (const float*)d_in[25];

  float* logits = (float*)d_out;          // output order: (logits, new_state)
  float* st_out = logits + VOC;

  float* ws    = (float*)d_ws;
  float* x     = ws;            // D
  float* kbuf  = ws + 1 * D;    // D
  float* vbuf  = ws + 2 * D;    // D
  float* rpre  = ws + 3 * D;    // D
  float* rpre2 = ws + 4 * D;    // D
  float* kkbuf = ws + 5 * D;    // H

  const dim3 blk(256);
  k_embed<<<1, blk, 0, stream>>>(token, emb, ln0_w, ln0_b, x);

  for (int l = 0; l < LNUM; ++l) {
    const float* st_l  = state  + (size_t)l * 5 * D;
    float*       sto_l = st_out + (size_t)l * 5 * D;
    const size_t mo = (size_t)l * D * D;  // D x D matrices
    k_att_qkv<<<dim3(D / RPB, 3), blk, 0, stream>>>(
        x, ln1_w + l * D, ln1_b + l * D, att_tmk + l * D, att_tmv + l * D,
        att_tmr + l * D, st_l + 1 * D, att_kw + mo, att_vw + mo, att_rw + mo,
        kbuf, vbuf, rpre, sto_l + 1 * D);
    k_att_out<<<dim3(D / RPB), blk, 0, stream>>>(
        x, kbuf, vbuf, rpre, st_l + 2 * D, st_l + 3 * D, st_l + 4 * D,
        time_first + l * D, time_decay + l * D, att_ow + mo,
        sto_l + 2 * D, sto_l + 3 * D, sto_l + 4 * D);
    k_ffn_rk<<<dim3(D / RPB + HD / RPB), blk, 0, stream>>>(
        x, ln2_w + l * D, ln2_b + l * D, ffn_tmk + l * D, ffn_tmr + l * D,
        st_l, ffn_rw + mo, ffn_kw + (size_t)l * HD * D, rpre2, kkbuf, sto_l);
    k_ffn_out<<<dim3(D / RPB), blk, 0, stream>>>(
        x, kkbuf, rpre2, ffn_vw + (size_t)l * D * HD);
  }

  k_head<<<dim3((VOC + RPB - 1) / RPB), blk, 0, stream>>>(
      x, lnout_w, lnout_b, head_w, logits);
}